// FrequencyAttention5_2250562863504
// MI455X (gfx1250) — compile-verified
//
#include <hip/hip_runtime.h>
#include <math.h>

typedef __attribute__((ext_vector_type(2))) float v2f;
typedef __attribute__((ext_vector_type(8))) float v8f;

// D = A(16x4) * B(4x16) + C  (f32 WMMA, wave32)
static __device__ __forceinline__ v8f wmma4(v2f a, v2f b, v8f c) {
  return __builtin_amdgcn_wmma_f32_16x16x4_f32(false, a, false, b, (short)0, c,
                                               false, false);
}

#define NQP   8388608    // 16384*512 floats
#define BATCH 32
#define SEQ   512
#define DIM   512
#define NFREQ 257
#define DPH   64
#define NH8   256        // B*H
#define SPITCH 516       // 512 + 4 pad (bank spread)

// ---------------------------------------------------------------------------
// GEMM: C[M,N] = X[M,K] @ W[K,N] + bias[N]
// 256 threads = 8 waves; block tile 64(M) x 512(N); each wave: 64x64 via
// 4x4 WMMA_F32_16x16x4 tiles, K stepped by 4.
// Inner loop: 8 fragment loads -> 16 WMMAs (2 flops-per-byte better than 2x2).
// ---------------------------------------------------------------------------
__global__ __launch_bounds__(256) void gemm_bias_kernel(
    const float* __restrict__ X, const float* __restrict__ W,
    const float* __restrict__ bias, float* __restrict__ C,
    int M, int N, int K) {
  const int lane = threadIdx.x & 31;
  const int wave = threadIdx.x >> 5;
  const int m0 = blockIdx.x * 64;
  const int n0 = blockIdx.y * 512 + wave * 64;
  const int lm  = lane & 15;         // M (or N) position within tile
  const int kh  = (lane >> 4) * 2;   // K offset: lanes 0-15 -> K{0,1}, 16-31 -> K{2,3}
  const int hi8 = (lane >> 4) * 8;   // C rows: lanes 16-31 hold M = 8 + i

  v8f acc[4][4];
#pragma unroll
  for (int i = 0; i < 4; ++i)
#pragma unroll
    for (int j = 0; j < 4; ++j)
      acc[i][j] = (v8f){0.f, 0.f, 0.f, 0.f, 0.f, 0.f, 0.f, 0.f};

  const float* ap0 = X + (size_t)(m0 + lm) * K + kh;
  const float* bp0 = W + (size_t)kh * N + n0 + lm;

  for (int k0 = 0; k0 < K; k0 += 4) {
    v2f a[4], b[4];
#pragma unroll
    for (int mt = 0; mt < 4; ++mt) {
      const float* ap = ap0 + (size_t)(mt * 16) * K + k0;
      a[mt].x = ap[0];
      a[mt].y = ap[1];
    }
#pragma unroll
    for (int nt = 0; nt < 4; ++nt) {
      const float* bp = bp0 + (size_t)k0 * N + nt * 16;
      b[nt].x = bp[0];
      b[nt].y = bp[N];
    }
#pragma unroll
    for (int mt = 0; mt < 4; ++mt)
#pragma unroll
      for (int nt = 0; nt < 4; ++nt)
        acc[mt][nt] = wmma4(a[mt], b[nt], acc[mt][nt]);
  }

#pragma unroll
  for (int mt = 0; mt < 4; ++mt)
#pragma unroll
    for (int nt = 0; nt < 4; ++nt) {
      const int col = n0 + nt * 16 + lm;
      const float bb = bias[col];
#pragma unroll
      for (int i = 0; i < 8; ++i) {
        const int row = m0 + mt * 16 + hi8 + i;
        C[(size_t)row * N + col] = acc[mt][nt][i] + bb;
      }
    }
}

// ---------------------------------------------------------------------------
// Forward DFT along L (rfft): Re[b,f,d] =  sum_l v[b,l,d] cos(2*pi*f*l/512)
//                             Im[b,f,d] = -sum_l v[b,l,d] sin(2*pi*f*l/512)
// grid (32, 33), 512 threads (d), 8 freqs per block via LDS trig table.
// ---------------------------------------------------------------------------
__global__ __launch_bounds__(512) void dft_fwd_kernel(
    const float* __restrict__ V, float* __restrict__ Re,
    float* __restrict__ Im) {
  __shared__ float ct[512], st[512];
  const int tid = threadIdx.x;
  {
    float ang = 6.2831853071795864769f * (float)tid / 512.0f;
    ct[tid] = cosf(ang);
    st[tid] = sinf(ang);
  }
  __syncthreads();

  const int b = blockIdx.x;
  const int f0 = blockIdx.y * 8;
  const float* Vb = V + (size_t)b * SEQ * DIM;

  float re[8], im[8];
#pragma unroll
  for (int j = 0; j < 8; ++j) { re[j] = 0.f; im[j] = 0.f; }

  for (int l = 0; l < SEQ; ++l) {
    const float vv = Vb[(size_t)l * DIM + tid];
#pragma unroll
    for (int j = 0; j < 8; ++j) {
      const int k = ((f0 + j) * l) & 511;
      re[j] += vv * ct[k];
      im[j] -= vv * st[k];
    }
  }
#pragma unroll
  for (int j = 0; j < 8; ++j) {
    const int f = f0 + j;
    if (f < NFREQ) {
      Re[((size_t)b * NFREQ + f) * DIM + tid] = re[j];
      Im[((size_t)b * NFREQ + f) * DIM + tid] = im[j];
    }
  }
}

// ---------------------------------------------------------------------------
// Top-K mask per (b,d): thresh = K-th largest amplitude (K=4); zero bins with
// amp < thresh; fold irfft weight w_f/N into surviving bins.
// K-th largest = min{ amp[i] : #{amp[j] > amp[i]} <= K-1 }  (tie-exact).
// One block per channel, 256 threads.
// ---------------------------------------------------------------------------
__global__ __launch_bounds__(256) void mask_kernel(float* __restrict__ Re,
                                                   float* __restrict__ Im) {
  __shared__ float amp[NFREQ];
  __shared__ float red[256];
  const int bd = blockIdx.x;
  const int b = bd >> 9;
  const int d = bd & 511;
  const int tid = threadIdx.x;

  for (int f = tid; f < NFREQ; f += 256) {
    const float re = Re[((size_t)b * NFREQ + f) * DIM + d];
    const float im = Im[((size_t)b * NFREQ + f) * DIM + d];
    amp[f] = sqrtf(re * re + im * im);
  }
  __syncthreads();

  float cand = 3.0e38f;
  for (int f = tid; f < NFREQ; f += 256) {
    const float a = amp[f];
    int c = 0;
    for (int j = 0; j < NFREQ; ++j) c += (amp[j] > a) ? 1 : 0;
    if (c <= 3) cand = fminf(cand, a);
  }
  red[tid] = cand;
  __syncthreads();
  for (int s = 128; s > 0; s >>= 1) {
    if (tid < s) red[tid] = fminf(red[tid], red[tid + s]);
    __syncthreads();
  }
  const float th = red[0];

  for (int f = tid; f < NFREQ; f += 256) {
    const float keepw =
        (amp[f] >= th)
            ? (((f == 0) || (f == NFREQ - 1)) ? (1.0f / 512.0f) : (2.0f / 512.0f))
            : 0.0f;
    Re[((size_t)b * NFREQ + f) * DIM + d] *= keepw;
    Im[((size_t)b * NFREQ + f) * DIM + d] *= keepw;
  }
}

// ---------------------------------------------------------------------------
// Inverse DFT: v'[b,l,d] = sum_f Re'[b,f,d] cos(th) - Im'[b,f,d] sin(th)
// (weights/normalization already folded). grid (32, 64), 512 threads, 8 l's.
// ---------------------------------------------------------------------------
__global__ __launch_bounds__(512) void dft_inv_kernel(
    const float* __restrict__ Re, const float* __restrict__ Im,
    float* __restrict__ Vout) {
  __shared__ float ct[512], st[512];
  const int tid = threadIdx.x;
  {
    float ang = 6.2831853071795864769f * (float)tid / 512.0f;
    ct[tid] = cosf(ang);
    st[tid] = sinf(ang);
  }
  __syncthreads();

  const int b = blockIdx.x;
  const int l0 = blockIdx.y * 8;

  float acc[8];
#pragma unroll
  for (int j = 0; j < 8; ++j) acc[j] = 0.f;

  for (int f = 0; f < NFREQ; ++f) {
    const float re = Re[((size_t)b * NFREQ + f) * DIM + tid];
    const float im = Im[((size_t)b * NFREQ + f) * DIM + tid];
#pragma unroll
    for (int j = 0; j < 8; ++j) {
      const int k = (f * (l0 + j)) & 511;
      acc[j] += re * ct[k] - im * st[k];
    }
  }
#pragma unroll
  for (int j = 0; j < 8; ++j)
    Vout[(size_t)b * SEQ * DIM + (size_t)(l0 + j) * DIM + tid] = acc[j];
}

// ---------------------------------------------------------------------------
// Attention over flat-reshaped heads: Q/K/V viewed as [256, 512, 64].
// Block = (n, qtile of 64 rows), 256 threads (8 waves).
// Phase 1: S(64x512) = scale * Q_tile K^T  (WMMA, scores to dynamic LDS)
// Phase 2: row softmax (4 threads per row)
// Phase 3: ctx(64x64) = P V                (WMMA, P fragments from LDS)
// ---------------------------------------------------------------------------
__global__ __launch_bounds__(256) void attn_kernel(
    const float* __restrict__ Qp, const float* __restrict__ Kp,
    const float* __restrict__ Vp, float* __restrict__ Ctx) {
  extern __shared__ float smem[];
  float* S = smem;                    // 64 x SPITCH
  float* red = smem + 64 * SPITCH;    // 256

  const int n = blockIdx.x;           // 0..255 (= b*H + h, flat view)
  const int q0 = blockIdx.y * 64;
  const int lane = threadIdx.x & 31;
  const int wave = threadIdx.x >> 5;
  const int lm = lane & 15;
  const int kh = (lane >> 4) * 2;
  const int hi8 = (lane >> 4) * 8;
  const float scale = 0.35355339059327373f;  // (DPH // H) ** -0.5 = 8^-0.5

  const float* Qb = Qp + (size_t)n * SEQ * DPH;
  const float* Kb = Kp + (size_t)n * SEQ * DPH;
  const float* Vb = Vp + (size_t)n * SEQ * DPH;

  // ---- Phase 1: scores. wave w covers key cols [w*64, w*64+64) ----
  for (int mt = 0; mt < 4; ++mt) {
    v2f a[16];
    const float* qrow = Qb + (size_t)(q0 + mt * 16 + lm) * DPH + kh;
#pragma unroll
    for (int ks = 0; ks < 16; ++ks) {
      a[ks].x = qrow[4 * ks];
      a[ks].y = qrow[4 * ks + 1];
    }
    for (int nt = 0; nt < 4; ++nt) {
      const int kr = wave * 64 + nt * 16 + lm;  // key row (= score col)
      const float* krow = Kb + (size_t)kr * DPH + kh;
      v8f c = (v8f){0.f, 0.f, 0.f, 0.f, 0.f, 0.f, 0.f, 0.f};
#pragma unroll
      for (int ks = 0; ks < 16; ++ks) {
        v2f bfr;
        bfr.x = krow[4 * ks];
        bfr.y = krow[4 * ks + 1];
        c = wmma4(a[ks], bfr, c);
      }
      const int col = wave * 64 + nt * 16 + lm;
#pragma unroll
      for (int i = 0; i < 8; ++i)
        S[(mt * 16 + hi8 + i) * SPITCH + col] = c[i] * scale;
    }
  }
  __syncthreads();

  // ---- Phase 2: softmax, 4 threads per row, 128 cols each ----
  {
    const int r = threadIdx.x >> 2;
    const int p = threadIdx.x & 3;
    float* Sr = S + r * SPITCH + p * 128;
    float m = -3.0e38f;
    for (int c = 0; c < 128; ++c) m = fmaxf(m, Sr[c]);
    red[r * 4 + p] = m;
    __syncthreads();
    m = fmaxf(fmaxf(red[r * 4 + 0], red[r * 4 + 1]),
              fmaxf(red[r * 4 + 2], red[r * 4 + 3]));
    __syncthreads();
    float sum = 0.f;
    for (int c = 0; c < 128; ++c) {
      const float e = __expf(Sr[c] - m);
      Sr[c] = e;
      sum += e;
    }
    red[r * 4 + p] = sum;
    __syncthreads();
    sum = red[r * 4 + 0] + red[r * 4 + 1] + red[r * 4 + 2] + red[r * 4 + 3];
    const float rinv = 1.0f / sum;
    for (int c = 0; c < 128; ++c) Sr[c] *= rinv;
  }
  __syncthreads();

  // ---- Phase 3: ctx = P @ V. wave -> (mt = w>>1, nt = 2*(w&1)..+1) ----
  {
    const int mt = wave >> 1;
    const int ntb = (wave & 1) * 2;
    v8f c0 = (v8f){0.f, 0.f, 0.f, 0.f, 0.f, 0.f, 0.f, 0.f};
    v8f c1 = c0;
    for (int k0 = 0; k0 < SEQ; k0 += 4) {
      v2f a;
      const float* sp = S + (mt * 16 + lm) * SPITCH + k0 + kh;
      a.x = sp[0];
      a.y = sp[1];
      const float* vrow = Vb + (size_t)(k0 + kh) * DPH;
      v2f b0, b1;
      b0.x = vrow[ntb * 16 + lm];
      b0.y = vrow[DPH + ntb * 16 + lm];
      b1.x = vrow[(ntb + 1) * 16 + lm];
      b1.y = vrow[DPH + (ntb + 1) * 16 + lm];
      c0 = wmma4(a, b0, c0);
      c1 = wmma4(a, b1, c1);
    }
    float* Cb = Ctx + (size_t)n * SEQ * DPH;
#pragma unroll
    for (int i = 0; i < 8; ++i) {
      const int row = q0 + mt * 16 + hi8 + i;
      Cb[(size_t)row * DPH + ntb * 16 + lm] = c0[i];
      Cb[(size_t)row * DPH + (ntb + 1) * 16 + lm] = c1[i];
    }
  }
}

// ---------------------------------------------------------------------------
// Residual add + LayerNorm over D=512. One block (256 threads) per row.
// ---------------------------------------------------------------------------
__global__ __launch_bounds__(256) void ln_kernel(
    const float* __restrict__ P, const float* __restrict__ Resid,
    const float* __restrict__ gamma, const float* __restrict__ beta,
    float* __restrict__ Out) {
  __shared__ float rsum[256], rsq[256];
  const int r = blockIdx.x;
  const int tid = threadIdx.x;
  const size_t base = (size_t)r * DIM;
  const float x0 = P[base + tid] + Resid[base + tid];
  const float x1 = P[base + 256 + tid] + Resid[base + 256 + tid];
  rsum[tid] = x0 + x1;
  rsq[tid] = x0 * x0 + x1 * x1;
  __syncthreads();
  for (int s = 128; s > 0; s >>= 1) {
    if (tid < s) {
      rsum[tid] += rsum[tid + s];
      rsq[tid] += rsq[tid + s];
    }
    __syncthreads();
  }
  const float mu = rsum[0] * (1.0f / 512.0f);
  const float var = rsq[0] * (1.0f / 512.0f) - mu * mu;
  const float rstd = rsqrtf(var + 1e-5f);
  Out[base + tid] = (x0 - mu) * rstd * gamma[tid] + beta[tid];
  Out[base + 256 + tid] = (x1 - mu) * rstd * gamma[256 + tid] + beta[256 + tid];
}

// ---------------------------------------------------------------------------
extern "C" void kernel_launch(void* const* d_in, const int* in_sizes, int n_in,
                              void* d_out, int out_size, void* d_ws,
                              size_t ws_size, hipStream_t stream) {
  const float* key   = (const float*)d_in[0];
  const float* value = (const float*)d_in[1];
  const float* query = (const float*)d_in[2];
  const float* Wk = (const float*)d_in[3];
  const float* bk = (const float*)d_in[4];
  const float* Wv = (const float*)d_in[5];
  const float* bv = (const float*)d_in[6];
  const float* Wq = (const float*)d_in[7];
  const float* bq = (const float*)d_in[8];
  const float* Wo = (const float*)d_in[9];
  const float* bo = (const float*)d_in[10];
  const float* gamma = (const float*)d_in[11];
  const float* beta  = (const float*)d_in[12];
  float* out = (float*)d_out;

  float* ws = (float*)d_ws;
  float* qp = ws;                         // 8388608 f
  float* kp = ws + (size_t)NQP;           // 8388608 f
  float* vp = ws + (size_t)2 * NQP;       // 8388608 f (later: filtered v)
  float* Re = ws + (size_t)3 * NQP;       // 4210688 f
  float* Im = Re + (size_t)BATCH * NFREQ * DIM;
  float* ctx = Re;                        // reuse Re+Im region after irfft
  float* proj = kp;                       // reuse kp after attention

  const dim3 gblk(256);
  const dim3 ggrid(16384 / 64, DIM / 512);  // (256, 1)

  gemm_bias_kernel<<<ggrid, gblk, 0, stream>>>(query, Wq, bq, qp, 16384, DIM, DIM);
  gemm_bias_kernel<<<ggrid, gblk, 0, stream>>>(key,   Wk, bk, kp, 16384, DIM, DIM);
  gemm_bias_kernel<<<ggrid, gblk, 0, stream>>>(value, Wv, bv, vp, 16384, DIM, DIM);

  dft_fwd_kernel<<<dim3(BATCH, 33), 512, 0, stream>>>(vp, Re, Im);
  mask_kernel<<<BATCH * DIM, 256, 0, stream>>>(Re, Im);
  dft_inv_kernel<<<dim3(BATCH, SEQ / 8), 512, 0, stream>>>(Re, Im, vp);

  const size_t attn_lds = (64 * SPITCH + 256) * sizeof(float);  // ~133 KB
  attn_kernel<<<dim3(NH8, SEQ / 64), 256, attn_lds, stream>>>(qp, kp, vp, ctx);

  gemm_bias_kernel<<<ggrid, gblk, 0, stream>>>(ctx, Wo, bo, proj, 16384, DIM, DIM);
  ln_kernel<<<BATCH * SEQ, 256, 0, stream>>>(proj, query, gamma, beta, out);
}